// MinibatchDiscrimination_33517924778715
// MI455X (gfx1250) — compile-verified
//
#include <hip/hip_runtime.h>
#include <hip/hip_bf16.h>

// ---- shapes from the reference ----
#define IN_F   512
#define OUT_F  64
#define INTER  16
#define BATCH  512
#define N_COLS (OUT_F * INTER)   // 1024
#define OUT_W  (IN_F + OUT_F)    // 576

typedef __attribute__((ext_vector_type(2))) float v2f;
typedef __attribute__((ext_vector_type(4))) float v4f;
typedef __attribute__((ext_vector_type(8))) float v8f;

// -----------------------------------------------------------------------------
// Phase 1: M = x @ T   (fp32 WMMA, V_WMMA_F32_16X16X4_F32)
// One wave (32 threads) computes one 16x16 tile of M. K loop: 512/4 = 128 steps.
//
// VGPR layouts (CDNA5 ISA 7.12.2):
//   A 16x4 f32 : lanes 0-15 -> row M=lane, VGPR0=K0, VGPR1=K1
//                lanes 16-31 -> row M=lane-16, VGPR0=K2, VGPR1=K3
//   B 4x16 f32 : lanes 0-15 -> col N=lane, VGPR0=K0-row, VGPR1=K1-row
//                lanes 16-31 -> col N=lane-16, VGPR0=K2-row, VGPR1=K3-row
//   C/D 16x16  : VGPR r -> row r (lanes 0-15), row r+8 (lanes 16-31)
// -----------------------------------------------------------------------------
__global__ void __launch_bounds__(32)
mbd_gemm_wmma(const float* __restrict__ x, const float* __restrict__ T,
              float* __restrict__ M) {
  const int nTile = blockIdx.x;      // 0..63  (N tiles of 16 over 1024)
  const int mTile = blockIdx.y;      // 0..31  (M tiles of 16 over 512)
  const int lane  = threadIdx.x;     // 0..31
  const int half  = lane >> 4;       // 0: K{0,1}, 1: K{2,3}
  const int l     = lane & 15;

  const int mBase = mTile * 16;
  const int nBase = nTile * 16;

  // Per-lane base pointers
  const float* aRow = x + (size_t)(mBase + l) * IN_F + 2 * half;   // x[row, k + 2*half]
  const float* bCol = T + (size_t)(2 * half) * N_COLS + nBase + l; // T[k + 2*half, n]

  v8f c = {};

#pragma unroll 4
  for (int k = 0; k < IN_F; k += 4) {
    // A: {x[row, k+2h], x[row, k+2h+1]}  (8B aligned: k%4==0)
    v2f a = *(const v2f*)(aRow + k);
    // B: {T[k+2h, n], T[k+2h+1, n]}
    v2f b;
    b.x = bCol[(size_t)k * N_COLS];
    b.y = bCol[(size_t)(k + 1) * N_COLS];
    c = __builtin_amdgcn_wmma_f32_16x16x4_f32(
        /*neg_a=*/false, a, /*neg_b=*/false, b,
        /*c_mod=*/(short)0, c, /*reuse_a=*/false, /*reuse_b=*/false);
  }

  // Store D: VGPR r -> rows (mBase + r + 8*half), col nBase + l
  float* dBase = M + (size_t)(mBase + 8 * half) * N_COLS + nBase + l;
#pragma unroll
  for (int r = 0; r < 8; ++r) {
    dBase[(size_t)r * N_COLS] = c[r];
  }
}

// -----------------------------------------------------------------------------
// Phase 2: for each (i, o): out[i, 512+o] = sum_j exp(-sum_k |M[i,o,k]-M[j,o,k]|) - 1
// Block: 256 threads; grid (2 i-blocks, 64 o). LDS stages M[:, o, :] = 32 KB.
// -----------------------------------------------------------------------------
__global__ void __launch_bounds__(256)
mbd_pairwise(const float* __restrict__ M, float* __restrict__ out) {
  __shared__ v4f sM[BATCH * 4];           // 512 rows x 16 floats = 32 KB

  const int o     = blockIdx.y;           // 0..63
  const int iBase = blockIdx.x * 256;     // 0 or 256
  const int tid   = threadIdx.x;

  // Cooperative load of column-block o: 2048 float4 / 256 threads = 8 each
  for (int t = tid; t < BATCH * 4; t += 256) {
    const int row = t >> 2;
    const int q   = t & 3;
    sM[t] = *(const v4f*)(M + (size_t)row * N_COLS + o * INTER + q * 4);
  }
  __syncthreads();

  const int i = iBase + tid;
  const v4f a0 = sM[i * 4 + 0];
  const v4f a1 = sM[i * 4 + 1];
  const v4f a2 = sM[i * 4 + 2];
  const v4f a3 = sM[i * 4 + 3];

  float acc = 0.0f;
#pragma unroll 4
  for (int j = 0; j < BATCH; ++j) {
    const v4f b0 = sM[j * 4 + 0];
    const v4f b1 = sM[j * 4 + 1];
    const v4f b2 = sM[j * 4 + 2];
    const v4f b3 = sM[j * 4 + 3];
    float l1 = fabsf(a0.x - b0.x) + fabsf(a0.y - b0.y) +
               fabsf(a0.z - b0.z) + fabsf(a0.w - b0.w);
    l1 += fabsf(a1.x - b1.x) + fabsf(a1.y - b1.y) +
          fabsf(a1.z - b1.z) + fabsf(a1.w - b1.w);
    l1 += fabsf(a2.x - b2.x) + fabsf(a2.y - b2.y) +
          fabsf(a2.z - b2.z) + fabsf(a2.w - b2.w);
    l1 += fabsf(a3.x - b3.x) + fabsf(a3.y - b3.y) +
          fabsf(a3.z - b3.z) + fabsf(a3.w - b3.w);
    acc += __expf(-l1);
  }

  // subtract the self term (exp(0) == 1) instead of masking in-loop
  out[(size_t)i * OUT_W + IN_F + o] = acc - 1.0f;
}

// -----------------------------------------------------------------------------
// Phase 3: out[i, 0:512] = x[i, :]
// -----------------------------------------------------------------------------
__global__ void __launch_bounds__(256)
mbd_copy_x(const float* __restrict__ x, float* __restrict__ out) {
  const int idx = blockIdx.x * 256 + threadIdx.x;   // 0 .. 512*512-1
  const int i = idx >> 9;
  const int c = idx & 511;
  out[(size_t)i * OUT_W + c] = x[idx];
}

extern "C" void kernel_launch(void* const* d_in, const int* in_sizes, int n_in,
                              void* d_out, int out_size, void* d_ws, size_t ws_size,
                              hipStream_t stream) {
  const float* x = (const float*)d_in[0];   // [512, 512]
  const float* T = (const float*)d_in[1];   // [512, 1024]
  float* out = (float*)d_out;               // [512, 576]
  float* M   = (float*)d_ws;                // [512, 1024] scratch (2 MB)

  // Phase 1: M = x @ T   (WMMA fp32; 64 N-tiles x 32 M-tiles, 1 wave each)
  mbd_gemm_wmma<<<dim3(64, 32), 32, 0, stream>>>(x, T, M);

  // Phase 3 can overlap with phase 2's dependency chain; launch early.
  mbd_copy_x<<<(BATCH * IN_F) / 256, 256, 0, stream>>>(x, out);

  // Phase 2: pairwise L1 + exp + masked sum (depends on M)
  mbd_pairwise<<<dim3(2, 64), 256, 0, stream>>>(M, out);
}